// KeyValueMemoryNetwork_14156212208280
// MI455X (gfx1250) — compile-verified
//
#include <hip/hip_runtime.h>
#include <hip/hip_bf16.h>
#include <math.h>

// ---------------------------------------------------------------------------
// KeyValueMemoryNetwork for MI455X (gfx1250, wave32).
//
//   B=4, S=512, E=128, fp32.
//   d_in[0]=key_seq[B,S,E]  d_in[1]=value_embed[B,S,S,E]  d_in[2]=hidden[B,S,E]
//   d_in[3]=mask[B,S,S]     d_in[4]=pos_emb[B,S,E]
//   d_out = dep_emb[B,S,E] ++ pos_out[B,S,E]
//   d_ws  = delta/key_score [B,S,S] fp32 (4 MB)
//
// Perf model: value_embed (512 MB, read once) dominates -> ~22us floor at
// 23.3 TB/s. GEMMs are ~0.8 GFLOP total -> use exact-precision
// V_WMMA_F32_16X16X4_F32 (fp32 WMMA): free at this scale, no precision loss
// before exp(). Streaming kernel uses non-temporal b128 loads so the
// read-once 512 MB doesn't evict useful L2 lines.
// ---------------------------------------------------------------------------

#define BB 4
#define SS 512
#define EE 128

typedef float v2f __attribute__((ext_vector_type(2)));
typedef float v4f __attribute__((ext_vector_type(4)));
typedef float v8f __attribute__((ext_vector_type(8)));

// ---------------------------------------------------------------------------
// Kernel 1: scores = hidden . key_seq^T / sqrt(E); delta = exp(scores)*mask
// One wave per 16x16 (q,k) tile. K-dim = E = 128, stepped 4 at a time with
// V_WMMA_F32_16X16X4_F32.
//
// fp32 WMMA fragment layout (wave32):
//   A 16x4 : lane L holds row m=L%16, k-pair kb=2*(L/16): {A[m][kb],A[m][kb+1]}
//   B 4x16 : lane L holds col n=L%16, k-pair kb=2*(L/16): {B[kb][n],B[kb+1][n]}
//   C 16x16: vgpr i, lane L -> row m=i+8*(L/16), col n=L%16
// ---------------------------------------------------------------------------
__global__ __launch_bounds__(256) void scores_delta_kernel(
    const float* __restrict__ hidden,    // [B,S,E]
    const float* __restrict__ key_seq,   // [B,S,E]
    const float* __restrict__ mask,      // [B,S,S]
    float* __restrict__ delta)           // [B,S,S]
{
    const int wave   = threadIdx.x >> 5;
    const int lane   = threadIdx.x & 31;
    const int tile   = blockIdx.x * 8 + wave;       // 4096 tiles total
    const int b      = tile >> 10;                  // / (32*32)
    const int qt     = (tile >> 5) & 31;
    const int kt     = tile & 31;

    const int ln16   = lane & 15;
    const int kb     = (lane >> 4) << 1;            // 0 or 2

    const float* Arow = hidden  + ((size_t)(b * SS + qt * 16 + ln16)) * EE;
    const float* Brow = key_seq + ((size_t)(b * SS + kt * 16 + ln16)) * EE;

    v8f c = {};
#pragma unroll
    for (int e0 = 0; e0 < EE; e0 += 4) {
        v2f a = *(const v2f*)(Arow + e0 + kb);      // 8B aligned
        v2f bf = *(const v2f*)(Brow + e0 + kb);
        c = __builtin_amdgcn_wmma_f32_16x16x4_f32(
                /*neg_a=*/false, a, /*neg_b=*/false, bf,
                /*c_mod=*/(short)0, c, /*reuse_a=*/false, /*reuse_b=*/false);
    }

    const float inv_scale = 0.088388347648318447f;  // 1/sqrt(128)
    const int   n   = kt * 16 + ln16;
    const int   mhi = (lane >> 4) << 3;             // 0 or 8
#pragma unroll
    for (int i = 0; i < 8; ++i) {
        const int q = qt * 16 + i + mhi;
        const size_t idx = ((size_t)(b * SS + q)) * SS + n;
        delta[idx] = expf(c[i] * inv_scale) * mask[idx];
    }
}

// ---------------------------------------------------------------------------
// Kernel 2: key_score = delta / (row_sum(delta) + 1e-10), in place.
// One 256-thread block per (b,q) row of 512.
// ---------------------------------------------------------------------------
__global__ __launch_bounds__(256) void normalize_kernel(float* __restrict__ ks)
{
    __shared__ float sred[256];
    const size_t base = (size_t)blockIdx.x * SS;
    const int t = threadIdx.x;

    const float v0 = ks[base + t];
    const float v1 = ks[base + t + 256];
    sred[t] = v0 + v1;
    __syncthreads();
#pragma unroll
    for (int s = 128; s > 0; s >>= 1) {
        if (t < s) sred[t] += sred[t + s];
        __syncthreads();
    }
    const float inv = 1.0f / (sred[0] + 1e-10f);
    ks[base + t]       = v0 * inv;
    ks[base + t + 256] = v1 * inv;
}

// ---------------------------------------------------------------------------
// Kernel 3: pos_out[b,q,e] = sum_k key_score[b,q,k] * pos_emb[b,k,e]
// One wave per 16(q) x 16(e) tile; K = 512 stepped by 4 (128 fp32 WMMAs).
// ---------------------------------------------------------------------------
__global__ __launch_bounds__(256) void pos_out_kernel(
    const float* __restrict__ ks,        // [B,S,S]
    const float* __restrict__ pos_emb,   // [B,S,E]
    float* __restrict__ pos_out)         // [B,S,E]
{
    const int wave = threadIdx.x >> 5;
    const int lane = threadIdx.x & 31;
    const int tile = blockIdx.x * 8 + wave;         // 4*32*8 = 1024 tiles
    const int b    = tile >> 8;                     // / (32*8)
    const int qt   = (tile >> 3) & 31;
    const int et   = tile & 7;

    const int ln16 = lane & 15;
    const int kb   = (lane >> 4) << 1;

    const float* Arow = ks + ((size_t)(b * SS + qt * 16 + ln16)) * SS;
    const float* Bcol = pos_emb + (size_t)b * SS * EE + et * 16 + ln16;

    v8f c = {};
    for (int k0 = 0; k0 < SS; k0 += 4) {
        v2f a = *(const v2f*)(Arow + k0 + kb);
        v2f bf;
        bf.x = Bcol[(size_t)(k0 + kb) * EE];
        bf.y = Bcol[(size_t)(k0 + kb + 1) * EE];
        c = __builtin_amdgcn_wmma_f32_16x16x4_f32(
                false, a, false, bf, (short)0, c, false, false);
    }

    const int n   = et * 16 + ln16;
    const int mhi = (lane >> 4) << 3;
#pragma unroll
    for (int i = 0; i < 8; ++i) {
        const int q = qt * 16 + i + mhi;
        pos_out[((size_t)(b * SS + q)) * EE + n] = c[i];
    }
}

// ---------------------------------------------------------------------------
// Kernel 4 (bandwidth-critical): dep_emb[b,q,e] = sum_k ks[b,q,k]*VE[b,q,k,e]
// One block per (b,q): 2048 blocks x 256 KB coalesced streaming.
// key_score row staged in LDS; 8 waves stride k; lane owns a float4 of e;
// non-temporal b128 loads (read-once data, keep L2 clean).
// ---------------------------------------------------------------------------
__global__ __launch_bounds__(256) void dep_emb_kernel(
    const float* __restrict__ ks,        // [B,S,S]
    const float* __restrict__ ve,        // [B,S,S,E]
    float* __restrict__ dep)             // [B,S,E]
{
    __shared__ float s_ks[SS];
    __shared__ v4f   s_red[8 * 32];

    const int row  = blockIdx.x;                    // b*S + q
    const int t    = threadIdx.x;
    const int wave = t >> 5;
    const int lane = t & 31;

    const float* ks_row = ks + (size_t)row * SS;
#pragma unroll
    for (int i = t; i < SS; i += 256) s_ks[i] = ks_row[i];
    __syncthreads();

    const float* ve_row = ve + (size_t)row * SS * EE + lane * 4;

    v4f acc = {};
#pragma unroll 4
    for (int k = wave; k < SS; k += 8) {
        const float w = s_ks[k];
        const v4f v = __builtin_nontemporal_load(
            (const v4f*)(ve_row + (size_t)k * EE));
        acc += w * v;
    }

    s_red[wave * 32 + lane] = acc;
    __syncthreads();

    if (t < 32) {
        v4f sum = {};
#pragma unroll
        for (int w = 0; w < 8; ++w) sum += s_red[w * 32 + t];
        *(v4f*)(dep + (size_t)row * EE + t * 4) = sum;
    }
}

// ---------------------------------------------------------------------------
extern "C" void kernel_launch(void* const* d_in, const int* in_sizes, int n_in,
                              void* d_out, int out_size, void* d_ws, size_t ws_size,
                              hipStream_t stream)
{
    const float* key_seq  = (const float*)d_in[0];
    const float* ve       = (const float*)d_in[1];
    const float* hidden   = (const float*)d_in[2];
    const float* mask     = (const float*)d_in[3];
    const float* pos_emb  = (const float*)d_in[4];

    float* dep_out = (float*)d_out;                       // [B,S,E]
    float* pos_out = (float*)d_out + (size_t)BB * SS * EE; // [B,S,E]
    float* delta   = (float*)d_ws;                        // [B,S,S] = 4 MB

    // 1) scores + exp*mask: 4096 tiles, 8 waves/block
    scores_delta_kernel<<<512, 256, 0, stream>>>(hidden, key_seq, mask, delta);
    // 2) in-place row normalize: one block per (b,q)
    normalize_kernel<<<BB * SS, 256, 0, stream>>>(delta);
    // 3) pos_out GEMM: 1024 tiles, 8 waves/block
    pos_out_kernel<<<128, 256, 0, stream>>>(delta, pos_emb, pos_out);
    // 4) value_embed streaming reduction (the 512 MB pass)
    dep_emb_kernel<<<BB * SS, 256, 0, stream>>>(delta, ve, dep_out);
}